// HDPHMM_70463233458518
// MI455X (gfx1250) — compile-verified
//
#include <hip/hip_runtime.h>

// ---------------------------------------------------------------------------
// HDP-HMM forward/backward, MI455X (gfx1250), wave32.
// Pipeline:
//   1) prep:     stick-breaking weights, row-softmax P (+ P^T), emission
//                weight matrix W(32x64), bias C(64), a0.
//   2) emission: E = exp(em) via f32 WMMA GEMM  [x^2 | x](T x 32) @ W(32x64)+C
//   3) chain:    per-chunk 64x64 matrix chain products (f32 WMMA), with
//                per-step max-rescaling (values ~1e-13 per step -> must rescale)
//   4) combine:  sequential scan over 1024 chunk matrices -> boundary vectors
//   5) replay:   2048 parallel blocks re-run the exact per-step normalized
//                recurrence inside each chunk, writing every alpha/beta row
//   6) ll:       log(sum(alpha[T-1]) + eps)
// ---------------------------------------------------------------------------

#define T_LEN   262144
#define K_ST    64
#define F_DIM   16
#define CHUNK   256
#define NCHUNK  (T_LEN / CHUNK)     // 1024
#define EPSV    1e-10f
#define LOG_2PI 1.8378770664093453f

// workspace offsets (in floats). total ~25.4M floats (~97 MB)
#define OFF_P       0u          // P row-major            (4096)
#define OFF_PT      4096u       // P^T row-major          (4096)
#define OFF_W       8192u       // emission weights 32x64 (2048)
#define OFF_CB      10240u      // emission bias          (64)
#define OFF_A0      10304u      // normalized alpha[0]    (64)
#define OFF_ENTRY_A 16384u      // fwd chunk entry vecs   (NCHUNK*64)
#define OFF_ENTRY_B 81920u      // bwd chunk entry vecs   (NCHUNK*64)
#define OFF_GA      262144u     // fwd chunk matrices     (NCHUNK*4096)
#define OFF_GB      4456448u    // bwd chunk matrices     (NCHUNK*4096)
#define OFF_E       8650752u    // E = exp(em), T x 64

typedef float v2f __attribute__((ext_vector_type(2)));
typedef float v8f __attribute__((ext_vector_type(8)));

// D(16x16,f32) = A(16x4,f32) x B(4x16,f32) + C   -- CDNA5 V_WMMA_F32_16X16X4_F32
__device__ __forceinline__ v8f wmma4(v2f a, v2f b, v8f c) {
#if __has_builtin(__builtin_amdgcn_wmma_f32_16x16x4_f32)
  return __builtin_amdgcn_wmma_f32_16x16x4_f32(false, a, false, b, (short)0, c,
                                               false, false);
#else
  // keep-alive fallback so a missing builtin is visible via histogram, not a
  // failed compile (wmma=0 in that case -> adjust next round)
  c[0] += a.x * b.x + a.y * b.y;
  return c;
#endif
}

__device__ __forceinline__ float wave_sum32(float v) {
#pragma unroll
  for (int off = 16; off > 0; off >>= 1) v += __shfl_xor(v, off, 32);
  return v;
}
__device__ __forceinline__ float wave_max32(float v) {
#pragma unroll
  for (int off = 16; off > 0; off >>= 1) v = fmaxf(v, __shfl_xor(v, off, 32));
  return v;
}

// ---------------------------------------------------------------------------
// 1) prep
// ---------------------------------------------------------------------------
__global__ __launch_bounds__(64) void hdp_prep_kernel(
    const float* __restrict__ obs, const float* __restrict__ blog,
    const float* __restrict__ pil, const float* __restrict__ means,
    const float* __restrict__ lvars, float* __restrict__ ws) {
  __shared__ float s_w[K_ST];
  __shared__ float s_red[K_ST];
  __shared__ float s_norm;
  const int k = threadIdx.x;

  if (k == 0) {  // stick breaking (K=64: trivial serial)
    float cum = 1.0f;
    for (int i = 0; i < K_ST; i++) {
      float b = 1.0f / (1.0f + expf(-blog[i]));
      s_w[i] = b * cum;
      cum *= (1.0f - b);
    }
  }
  __syncthreads();

  // row-softmax of pi_logits -> P and P^T
  {
    float m = -1e30f;
    for (int j = 0; j < K_ST; j++) m = fmaxf(m, pil[k * K_ST + j]);
    float ssum = 0.0f;
    for (int j = 0; j < K_ST; j++) ssum += expf(pil[k * K_ST + j] - m);
    float inv = 1.0f / ssum;
    for (int j = 0; j < K_ST; j++) {
      float p = expf(pil[k * K_ST + j] - m) * inv;
      ws[OFF_P + k * K_ST + j] = p;
      ws[OFF_PT + j * K_ST + k] = p;
    }
  }

  // emission weights: em[t,k] = sum_f x^2*(-.5*iv) + sum_f x*(mu*iv) + C[k]
  float lvsum = 0.0f, msq = 0.0f, quad0 = 0.0f;
  for (int f = 0; f < F_DIM; f++) {
    float lv = lvars[k * F_DIM + f];
    float iv = expf(-lv);
    float mu = means[k * F_DIM + f];
    ws[OFF_W + f * K_ST + k] = -0.5f * iv;            // rows 0..15  <- x^2
    ws[OFF_W + (F_DIM + f) * K_ST + k] = mu * iv;     // rows 16..31 <- x
    lvsum += lv;
    msq += mu * mu * iv;
    float d = obs[f] - mu;  // obs row 0, for a0
    quad0 += d * d * iv;
  }
  ws[OFF_CB + k] = -0.5f * ((float)F_DIM * LOG_2PI + lvsum + msq);

  // a0 = normalize(beta_weights * exp(em[0]))
  float em0 = -0.5f * ((float)F_DIM * LOG_2PI + lvsum + quad0);
  float a = s_w[k] * expf(em0);
  s_red[k] = a;
  __syncthreads();
  if (k == 0) {
    float s = 0.0f;
    for (int i = 0; i < K_ST; i++) s += s_red[i];
    s_norm = 1.0f / (s + EPSV);
  }
  __syncthreads();
  ws[OFF_A0 + k] = a * s_norm;
}

// ---------------------------------------------------------------------------
// 2) emission GEMM: 8 waves/block, each wave a 16x64 strip. 32 WMMA per wave.
// ---------------------------------------------------------------------------
__global__ __launch_bounds__(256) void hdp_emission_kernel(
    const float* __restrict__ obs, float* __restrict__ ws) {
  __shared__ float sW[32 * 64];
  __shared__ float sCb[64];
  __shared__ float sX[128 * 17];  // 128 obs rows, padded stride 17 (bank-safe)

  const int tid = threadIdx.x;
  float* Eo = ws + OFF_E;

  for (int i = tid; i < 32 * 64; i += 256) sW[i] = ws[OFF_W + i];
  if (tid < 64) sCb[tid] = ws[OFF_CB + tid];
  const int rowBase = blockIdx.x * 128;
  for (int i = tid; i < 128 * 16; i += 256) {
    int r = i >> 4, c = i & 15;
    sX[r * 17 + c] = obs[(size_t)(rowBase + r) * F_DIM + c];
  }
  __syncthreads();

  const int wave = tid >> 5, l = tid & 31, hi = l >> 4, lr = l & 15;
  const int arow = (wave * 16 + lr) * 17;

  const v8f zero = {0, 0, 0, 0, 0, 0, 0, 0};
  v8f acc[4];
#pragma unroll
  for (int jt = 0; jt < 4; jt++) acc[jt] = zero;

#pragma unroll
  for (int kc = 0; kc < 8; kc++) {  // GEMM K = 32, chunks of 4
    const int c0 = kc * 4 + hi * 2;  // A cols (per ISA 16x4 f32 layout)
    const int cc = c0 & 15;
    float vx = sX[arow + cc], vy = sX[arow + cc + 1];
    v2f a;
    if (kc < 4) { a.x = vx * vx; a.y = vy * vy; }   // x^2 half
    else        { a.x = vx;      a.y = vy;      }   // x half
#pragma unroll
    for (int jt = 0; jt < 4; jt++) {
      const int n = jt * 16 + lr;
      v2f b;
      b.x = sW[c0 * 64 + n];
      b.y = sW[(c0 + 1) * 64 + n];
      acc[jt] = wmma4(a, b, acc[jt]);
    }
  }

#pragma unroll
  for (int jt = 0; jt < 4; jt++) {
    const int col = jt * 16 + lr;
    const float cb = sCb[col];
#pragma unroll
    for (int g = 0; g < 8; g++) {
      const int row = rowBase + wave * 16 + g + hi * 8;  // D layout: m=g(+8)
      Eo[(size_t)row * 64 + col] = expf(acc[jt][g] + cb);
    }
  }
}

// ---------------------------------------------------------------------------
// 3) chunk chain products. dir 0: G <- diag(e_t) * (P^T G)   (forward)
//                          dir 1: H <- P * (diag(e_{t+1}) H) (backward)
// 4 waves, each owns a 16-row strip of the 64x64 product: 64 WMMA / step.
// ---------------------------------------------------------------------------
__global__ __launch_bounds__(128) void hdp_chain_kernel(float* __restrict__ ws) {
  const int dir = blockIdx.y;
  const int cid = blockIdx.x;
  const int tid = threadIdx.x;
  const int wave = tid >> 5, l = tid & 31, hi = l >> 4, lr = l & 15;

  __shared__ float sM[64 * 65];
  __shared__ float sG[2][64 * 65];
  __shared__ float sE[64];
  __shared__ float sRed[4];
  __shared__ float sInv;

  const float* Mg = ws + (dir == 0 ? OFF_PT : OFF_P);
  const float* Eg = ws + OFF_E;

  for (int i = tid; i < 4096; i += 128) {
    int r = i >> 6, cc = i & 63;
    sM[r * 65 + cc] = Mg[i];
    sG[0][r * 65 + cc] = (r == cc) ? 1.0f : 0.0f;
  }
  __syncthreads();

  int tfirst, nsteps, tinc;
  if (dir == 0) {
    if (cid == 0) { tfirst = 1; nsteps = CHUNK - 1; }
    else          { tfirst = cid * CHUNK; nsteps = CHUNK; }
    tinc = 1;
  } else {
    if (cid == NCHUNK - 1) { tfirst = T_LEN - 2; nsteps = CHUNK - 1; }
    else                   { tfirst = (cid + 1) * CHUNK - 1; nsteps = CHUNK; }
    tinc = -1;
  }

  const v8f zero = {0, 0, 0, 0, 0, 0, 0, 0};
  const int arow = (wave * 16 + lr) * 65;
  int cur = 0;

  for (int s = 0; s < nsteps; s++) {
    const int t = tfirst + s * tinc;
    const int eidx = (dir == 0) ? t : (t + 1);
    if (tid < 64) sE[tid] = Eg[(size_t)eidx * 64 + tid];
    __syncthreads();

    v8f acc[4];
#pragma unroll
    for (int jt = 0; jt < 4; jt++) acc[jt] = zero;

    const float* Gc = sG[cur];
#pragma unroll
    for (int kc = 0; kc < 16; kc++) {  // K = 64, chunks of 4
      const int k0 = kc * 4 + hi * 2;
      v2f a;
      a.x = sM[arow + k0];
      a.y = sM[arow + k0 + 1];
      float es0 = 1.0f, es1 = 1.0f;
      if (dir == 1) { es0 = sE[k0]; es1 = sE[k0 + 1]; }
#pragma unroll
      for (int jt = 0; jt < 4; jt++) {
        const int n = jt * 16 + lr;
        v2f b;
        b.x = Gc[k0 * 65 + n] * es0;
        b.y = Gc[(k0 + 1) * 65 + n] * es1;
        acc[jt] = wmma4(a, b, acc[jt]);
      }
    }

    // forward: scale output rows by e_t; then rescale by global max (values
    // are all >= 0) so the 256-long product never underflows.
    float lm = 0.0f;
#pragma unroll
    for (int jt = 0; jt < 4; jt++) {
#pragma unroll
      for (int g = 0; g < 8; g++) {
        float v = acc[jt][g];
        if (dir == 0) v *= sE[wave * 16 + g + hi * 8];
        acc[jt][g] = v;
        lm = fmaxf(lm, v);
      }
    }
    lm = wave_max32(lm);
    if (l == 0) sRed[wave] = lm;
    __syncthreads();
    if (tid == 0)
      sInv = 1.0f / fmaxf(fmaxf(fmaxf(sRed[0], sRed[1]),
                                fmaxf(sRed[2], sRed[3])), 1e-30f);
    __syncthreads();
    const float inv = sInv;
    float* Gn = sG[cur ^ 1];
#pragma unroll
    for (int jt = 0; jt < 4; jt++) {
      const int n = jt * 16 + lr;
#pragma unroll
      for (int g = 0; g < 8; g++)
        Gn[(wave * 16 + g + hi * 8) * 65 + n] = acc[jt][g] * inv;
    }
    __syncthreads();
    cur ^= 1;
  }

  float* Gout = ws + (dir == 0 ? OFF_GA : OFF_GB) + (size_t)cid * 4096;
  for (int i = tid; i < 4096; i += 128) {
    int r = i >> 6, cc = i & 63;
    Gout[i] = sG[cur][r * 65 + cc];
  }
}

// ---------------------------------------------------------------------------
// 4) sequential combine across chunks (block 0: fwd, block 1: bwd)
// ---------------------------------------------------------------------------
__global__ __launch_bounds__(64) void hdp_combine_kernel(float* __restrict__ ws) {
  const int dir = blockIdx.x;
  const int i = threadIdx.x;
  __shared__ float sGt[64 * 65];   // transposed, padded (bank-safe)
  __shared__ float sv[64];
  __shared__ float sred[64];
  __shared__ float sinv;

  const float* Gbase = ws + (dir == 0 ? OFF_GA : OFF_GB);
  float* entry = ws + (dir == 0 ? OFF_ENTRY_A : OFF_ENTRY_B);

  sv[i] = (dir == 0) ? ws[OFF_A0 + i] : 1.0f;
  entry[(size_t)(dir == 0 ? 0 : NCHUNK - 1) * 64 + i] = sv[i];
  __syncthreads();

  for (int s = 0; s < NCHUNK - 1; s++) {
    const int c = (dir == 0) ? s : (NCHUNK - 1 - s);
    const float* G = Gbase + (size_t)c * 4096;
    for (int r = 0; r < 64; r++) sGt[i * 65 + r] = G[r * 64 + i];  // coalesced
    __syncthreads();
    float acc = 0.0f;
    for (int k = 0; k < 64; k++) acc += sGt[k * 65 + i] * sv[k];   // G[i][k]
    sred[i] = acc;
    __syncthreads();
    if (i == 0) {
      float s2 = 0.0f;
      for (int j = 0; j < 64; j++) s2 += sred[j];
      sinv = 1.0f / (s2 + EPSV);
    }
    __syncthreads();
    sv[i] = acc * sinv;
    __syncthreads();
    const int cn = (dir == 0) ? (c + 1) : (c - 1);
    entry[(size_t)cn * 64 + i] = sv[i];
  }
}

// ---------------------------------------------------------------------------
// 5) replay: one wave per chunk, exact normalized recurrence per step
// ---------------------------------------------------------------------------
__global__ __launch_bounds__(32) void hdp_replay_kernel(
    const float* __restrict__ ws, float* __restrict__ out) {
  const int dir = blockIdx.y;
  const int cid = blockIdx.x;
  const int l = threadIdx.x;
  const int j0 = 2 * l;

  __shared__ float sM[64 * 65];
  __shared__ float sa[64];
  __shared__ float swv[64];

  // fwd: a_new[j] = sum_i a[i] P[i][j]     -> sM = P (row-major)
  // bwd: b_new[i] = sum_j P[i][j] w[j]     -> sM = P^T (row-major)
  const float* Mg = ws + (dir == 0 ? OFF_P : OFF_PT);
  for (int i = l; i < 4096; i += 32) {
    int r = i >> 6, cc = i & 63;
    sM[r * 65 + cc] = Mg[i];
  }
  const float* entry = ws + (dir == 0 ? OFF_ENTRY_A : OFF_ENTRY_B) + (size_t)cid * 64;
  sa[j0] = entry[j0];
  sa[j0 + 1] = entry[j0 + 1];
  const float* E = ws + OFF_E;
  float* ob = out + (dir == 0 ? (size_t)0 : (size_t)T_LEN * 64);
  __syncthreads();

  if (dir == 0) {
    int tstart, nst;
    if (cid == 0) {
      ob[j0] = sa[j0]; ob[j0 + 1] = sa[j0 + 1];  // alpha[0] = a0
      tstart = 1; nst = CHUNK - 1;
    } else { tstart = cid * CHUNK; nst = CHUNK; }
    for (int s = 0; s < nst; s++) {
      const int t = tstart + s;
      const float e0 = E[(size_t)t * 64 + j0];
      const float e1 = E[(size_t)t * 64 + j0 + 1];
      float a0 = 0.0f, a1 = 0.0f;
      for (int i = 0; i < 64; i++) {
        const float av = sa[i];
        a0 += av * sM[i * 65 + j0];
        a1 += av * sM[i * 65 + j0 + 1];
      }
      a0 *= e0; a1 *= e1;
      const float inv = 1.0f / (wave_sum32(a0 + a1) + EPSV);
      a0 *= inv; a1 *= inv;
      ob[(size_t)t * 64 + j0] = a0;
      ob[(size_t)t * 64 + j0 + 1] = a1;
      __syncthreads();
      sa[j0] = a0; sa[j0 + 1] = a1;
      __syncthreads();
    }
  } else {
    int tstart, nst;
    if (cid == NCHUNK - 1) {
      ob[(size_t)(T_LEN - 1) * 64 + j0] = 1.0f;     // beta[T-1] = ones
      ob[(size_t)(T_LEN - 1) * 64 + j0 + 1] = 1.0f;
      tstart = T_LEN - 2; nst = CHUNK - 1;
    } else { tstart = (cid + 1) * CHUNK - 1; nst = CHUNK; }
    for (int s = 0; s < nst; s++) {
      const int t = tstart - s;
      const float e0 = E[(size_t)(t + 1) * 64 + j0];
      const float e1 = E[(size_t)(t + 1) * 64 + j0 + 1];
      swv[j0] = sa[j0] * e0;
      swv[j0 + 1] = sa[j0 + 1] * e1;
      __syncthreads();
      float b0 = 0.0f, b1 = 0.0f;
      for (int j = 0; j < 64; j++) {
        const float wv = swv[j];
        b0 += wv * sM[j * 65 + j0];       // P^T[j][i] = P[i][j]
        b1 += wv * sM[j * 65 + j0 + 1];
      }
      const float inv = 1.0f / (wave_sum32(b0 + b1) + EPSV);
      b0 *= inv; b1 *= inv;
      ob[(size_t)t * 64 + j0] = b0;
      ob[(size_t)t * 64 + j0 + 1] = b1;
      __syncthreads();
      sa[j0] = b0; sa[j0 + 1] = b1;
      __syncthreads();
    }
  }
}

// ---------------------------------------------------------------------------
// 6) log-likelihood
// ---------------------------------------------------------------------------
__global__ void hdp_ll_kernel(float* __restrict__ out) {
  if (threadIdx.x == 0) {
    const float* last = out + (size_t)(T_LEN - 1) * 64;
    float s = 0.0f;
    for (int k = 0; k < 64; k++) s += last[k];
    out[(size_t)2 * T_LEN * 64] = logf(s + EPSV);
  }
}

// ---------------------------------------------------------------------------
extern "C" void kernel_launch(void* const* d_in, const int* in_sizes, int n_in,
                              void* d_out, int out_size, void* d_ws,
                              size_t ws_size, hipStream_t stream) {
  (void)in_sizes; (void)n_in; (void)out_size; (void)ws_size;
  const float* obs   = (const float*)d_in[0];
  const float* blog  = (const float*)d_in[1];
  const float* pil   = (const float*)d_in[2];
  const float* means = (const float*)d_in[3];
  const float* lvars = (const float*)d_in[4];
  float* out = (float*)d_out;
  float* ws  = (float*)d_ws;   // needs ~97 MB

  hdp_prep_kernel<<<1, 64, 0, stream>>>(obs, blog, pil, means, lvars, ws);
  hdp_emission_kernel<<<T_LEN / 128, 256, 0, stream>>>(obs, ws);
  hdp_chain_kernel<<<dim3(NCHUNK, 2), 128, 0, stream>>>(ws);
  hdp_combine_kernel<<<2, 64, 0, stream>>>(ws);
  hdp_replay_kernel<<<dim3(NCHUNK, 2), 32, 0, stream>>>(ws, out);
  hdp_ll_kernel<<<1, 32, 0, stream>>>(out);
}